// GCNN_4329327034491
// MI455X (gfx1250) — compile-verified
//
#include <hip/hip_runtime.h>
#include <hip/hip_bf16.h>

typedef float v2f __attribute__((ext_vector_type(2)));
typedef float v8f __attribute__((ext_vector_type(8)));

#define BN_EPS 1e-5f

// ---------------------------------------------------------------- zero fill
__global__ void zero_f32(float* __restrict__ p, int n) {
  int i = blockIdx.x * blockDim.x + threadIdx.x;
  if (i < n) p[i] = 0.0f;
}

// --------------------------------------------------- edge scatter (agg add)
// agg[dst[e]] += x[src[e]] * ew[e]; one thread = one edge * 4 features
__global__ void scatter_edges(const float* __restrict__ x, const int* __restrict__ src,
                              const int* __restrict__ dst, const float* __restrict__ ew,
                              float* __restrict__ agg, int E, int F, int pfShift) {
  int idx = blockIdx.x * blockDim.x + threadIdx.x;
  int total = E << pfShift;               // E * F/4
  if (idx >= total) return;
  int e = idx >> pfShift;
  int f = (idx & ((1 << pfShift) - 1)) << 2;
  int s = src[e];
  int d = dst[e];
  float w = ew[e];
  const float4 v = *reinterpret_cast<const float4*>(x + (size_t)s * F + f);
  float* a = agg + (size_t)d * F + f;
  unsafeAtomicAdd(a + 0, v.x * w);
  unsafeAtomicAdd(a + 1, v.y * w);
  unsafeAtomicAdd(a + 2, v.z * w);
  unsafeAtomicAdd(a + 3, v.w * w);
}

// ------------------------------------------------- fused dual-GEMM + epilog
// Y = epilogue( Agg @ Wl + Xin @ Wr + bias )
// mode 0: relu ; mode 1: BN(eval) then relu
// block = 256 threads (8 waves, 4x2), tile 64x64, KB = 32, WMMA f32 16x16x4
// B tile stored K-pair interleaved so each WMMA B fragment is a contiguous
// 8-byte pair in LDS (single ds_load_b64, no register shuffles).
__launch_bounds__(256)
__global__ void graphconv_gemm(const float* __restrict__ Agg, const float* __restrict__ Xin,
                               const float* __restrict__ Wl, const float* __restrict__ Wr,
                               const float* __restrict__ bias,
                               const float* __restrict__ bn_g, const float* __restrict__ bn_b,
                               const float* __restrict__ bn_m, const float* __restrict__ bn_v,
                               float* __restrict__ Y, int N, int K, int M, int mode) {
  __shared__ float As[64][36];     // 64 rows x 32 cols (+pad, 16B-aligned rows)
  __shared__ float BsP[16][132];   // 16 K-pairs x (64 cols x 2 interleaved) (+pad)

  const int tid  = threadIdx.x;
  const int lane = tid & 31;
  const int wave = tid >> 5;
  const int wy   = wave >> 1;      // 0..3  -> 16-row strip
  const int wx   = wave & 1;       // 0..1  -> 32-col strip
  const int half = lane >> 4;      // K-pair select (f32 WMMA A/B layout)
  const int l16  = lane & 15;

  const int rowBase = blockIdx.y * 64;
  const int colBase = blockIdx.x * 64;

  v8f acc0 = {0.f, 0.f, 0.f, 0.f, 0.f, 0.f, 0.f, 0.f};
  v8f acc1 = {0.f, 0.f, 0.f, 0.f, 0.f, 0.f, 0.f, 0.f};

  // loader indices (B tile): 2x4 block per thread, one pass = full 32x64 tile
  const int brp = tid >> 4;          // K-pair row 0..15
  const int bcq = (tid & 15) << 2;   // col 0..60 step 4

  for (int pass = 0; pass < 2; ++pass) {
    const float* __restrict__ Ap = pass ? Xin : Agg;
    const float* __restrict__ Wp = pass ? Wr  : Wl;
    for (int kb = 0; kb < K; kb += 32) {
      __syncthreads();
      // ---- A tile: 64x32, 8 float4 per row, 2 per thread
#pragma unroll
      for (int i = 0; i < 2; ++i) {
        int t = tid + i * 256;
        int r = t >> 3;
        int c = (t & 7) << 2;
        int gr = rowBase + r;
        float4 v = make_float4(0.f, 0.f, 0.f, 0.f);
        if (gr < N) v = *reinterpret_cast<const float4*>(Ap + (size_t)gr * K + kb + c);
        *reinterpret_cast<float4*>(&As[r][c]) = v;
      }
      // ---- B tile: rows kb+2*brp, kb+2*brp+1, cols colBase+bcq..+3,
      //      stored K-pair interleaved: BsP[brp][c*2 + (k&1)]
      {
        const float* w0 = Wp + (size_t)(kb + 2 * brp) * M + colBase + bcq;
        float4 a0 = *reinterpret_cast<const float4*>(w0);
        float4 a1 = *reinterpret_cast<const float4*>(w0 + M);
        float* bp = &BsP[brp][bcq << 1];
        reinterpret_cast<float2*>(bp)[0] = make_float2(a0.x, a1.x);
        reinterpret_cast<float2*>(bp)[1] = make_float2(a0.y, a1.y);
        reinterpret_cast<float2*>(bp)[2] = make_float2(a0.z, a1.z);
        reinterpret_cast<float2*>(bp)[3] = make_float2(a0.w, a1.w);
      }
      __syncthreads();
      const int ar = wy * 16 + l16;
      const int bc0 = (wx * 32 + l16) << 1;
      const int bc1 = (wx * 32 + 16 + l16) << 1;
#pragma unroll
      for (int k = 0; k < 32; k += 4) {
        int kp = (k >> 1) + half;   // K-pair index for this half-wave
        v2f a  = *reinterpret_cast<const v2f*>(&As[ar][k + 2 * half]);
        v2f b0 = *reinterpret_cast<const v2f*>(&BsP[kp][bc0]);
        v2f b1 = *reinterpret_cast<const v2f*>(&BsP[kp][bc1]);
        acc0 = __builtin_amdgcn_wmma_f32_16x16x4_f32(false, a, false, b0, (short)0, acc0, false, false);
        acc1 = __builtin_amdgcn_wmma_f32_16x16x4_f32(false, a, false, b1, (short)0, acc1, false, false);
      }
    }
  }

  // epilogue: bias (+ folded BN) + relu, store
  const int gc0 = colBase + wx * 32 + l16;
  const int gc1 = gc0 + 16;
  float s0 = 1.f, s1 = 1.f;
  float t0 = bias[gc0], t1 = bias[gc1];
  if (mode == 1) {
    float sc0 = bn_g[gc0] * rsqrtf(bn_v[gc0] + BN_EPS);
    float sc1 = bn_g[gc1] * rsqrtf(bn_v[gc1] + BN_EPS);
    t0 = (t0 - bn_m[gc0]) * sc0 + bn_b[gc0];
    t1 = (t1 - bn_m[gc1]) * sc1 + bn_b[gc1];
    s0 = sc0; s1 = sc1;
  }
#pragma unroll
  for (int r = 0; r < 8; ++r) {
    int gr = rowBase + wy * 16 + 8 * half + r;
    if (gr < N) {
      float v0 = fmaxf(acc0[r] * s0 + t0, 0.f);
      float v1 = fmaxf(acc1[r] * s1 + t1, 0.f);
      Y[(size_t)gr * M + gc0] = v0;
      Y[(size_t)gr * M + gc1] = v1;
    }
  }
}

// --------------------------------------------------------- global mean pool
__global__ void pool_sum(const float* __restrict__ h, const int* __restrict__ batch,
                         float* __restrict__ gsum, int N) {
  int idx = blockIdx.x * blockDim.x + threadIdx.x;   // N * (256/4)
  if (idx >= (N << 6)) return;
  int n = idx >> 6;
  int f = (idx & 63) << 2;
  int b = batch[n];
  const float4 v = *reinterpret_cast<const float4*>(h + (size_t)n * 256 + f);
  float* g = gsum + (size_t)b * 256 + f;
  unsafeAtomicAdd(g + 0, v.x);
  unsafeAtomicAdd(g + 1, v.y);
  unsafeAtomicAdd(g + 2, v.z);
  unsafeAtomicAdd(g + 3, v.w);
}

__global__ void pool_cnt(const int* __restrict__ batch, float* __restrict__ gcnt, int N) {
  int n = blockIdx.x * blockDim.x + threadIdx.x;
  if (n < N) unsafeAtomicAdd(&gcnt[batch[n]], 1.0f);
}

// --------------------------------------------------------------- MLP head
// one block per graph (64 threads): relu(g@W4+b4) @ W5 + b5, relu
__global__ void head_mlp(const float* __restrict__ gsum, const float* __restrict__ gcnt,
                         const float* __restrict__ W4, const float* __restrict__ b4,
                         const float* __restrict__ W5, const float* __restrict__ b5,
                         float* __restrict__ out) {
  int g = blockIdx.x;
  int j = threadIdx.x;   // 0..63
  float inv = 1.0f / fmaxf(gcnt[g], 1.0f);
  float h = b4[j];
  const float* gs = gsum + (size_t)g * 256;
  for (int k = 0; k < 256; ++k) h += gs[k] * inv * W4[k * 64 + j];
  h = fmaxf(h, 0.f);
  __shared__ float red[64];
  red[j] = h * W5[j];
  __syncthreads();
  for (int s = 32; s > 0; s >>= 1) {
    if (j < s) red[j] += red[j + s];
    __syncthreads();
  }
  if (j == 0) out[g] = fmaxf(red[0] + b5[0], 0.f);
}

// ---------------------------------------------------------------- launcher
extern "C" void kernel_launch(void* const* d_in, const int* in_sizes, int n_in,
                              void* d_out, int out_size, void* d_ws, size_t ws_size,
                              hipStream_t stream) {
  const float* x     = (const float*)d_in[0];
  const int*   eidx  = (const int*)d_in[1];
  const float* ew    = (const float*)d_in[2];
  const int*   batch = (const int*)d_in[3];
  const float* Wl1 = (const float*)d_in[4];
  const float* bl1 = (const float*)d_in[5];
  const float* Wr1 = (const float*)d_in[6];
  const float* Wl2 = (const float*)d_in[7];
  const float* bl2 = (const float*)d_in[8];
  const float* Wr2 = (const float*)d_in[9];
  const float* Wl3 = (const float*)d_in[10];
  const float* bl3 = (const float*)d_in[11];
  const float* Wr3 = (const float*)d_in[12];
  const float* bng = (const float*)d_in[13];
  const float* bnb = (const float*)d_in[14];
  const float* bnm = (const float*)d_in[15];
  const float* bnv = (const float*)d_in[16];
  const float* W4  = (const float*)d_in[17];
  const float* b4  = (const float*)d_in[18];
  const float* W5  = (const float*)d_in[19];
  const float* b5  = (const float*)d_in[20];

  const int E = in_sizes[2];      // 800000
  const int N = in_sizes[3];      // 50000
  const int* src = eidx;
  const int* dst = eidx + E;

  float* ws   = (float*)d_ws;
  float* agg  = ws;                                // N*512 (layer1 uses N*128)
  float* h1   = ws + (size_t)N * 512;              // N*512 (reused as h3: N*256)
  float* h2   = ws + (size_t)N * 512 * 2;          // N*512
  float* gsum = ws + (size_t)N * 512 * 3;          // 64*256
  float* gcnt = gsum + 64 * 256;                   // 64
  float* h3   = h1;

  auto nb = [](int n, int b) { return (n + b - 1) / b; };
  const int rowsBlk = (N + 63) / 64;

  // ---- layer 1: 128 -> 512, relu
  zero_f32<<<nb(N * 128, 256), 256, 0, stream>>>(agg, N * 128);
  scatter_edges<<<nb(E * 32, 256), 256, 0, stream>>>(x, src, dst, ew, agg, E, 128, 5);
  graphconv_gemm<<<dim3(512 / 64, rowsBlk), 256, 0, stream>>>(
      agg, x, Wl1, Wr1, bl1, bng, bnb, bnm, bnv, h1, N, 128, 512, 0);

  // ---- layer 2: 512 -> 512, BN + relu
  zero_f32<<<nb(N * 512, 256), 256, 0, stream>>>(agg, N * 512);
  scatter_edges<<<nb(E * 128, 256), 256, 0, stream>>>(h1, src, dst, ew, agg, E, 512, 7);
  graphconv_gemm<<<dim3(512 / 64, rowsBlk), 256, 0, stream>>>(
      agg, h1, Wl2, Wr2, bl2, bng, bnb, bnm, bnv, h2, N, 512, 512, 1);

  // ---- layer 3: 512 -> 256, relu
  zero_f32<<<nb(N * 512, 256), 256, 0, stream>>>(agg, N * 512);
  scatter_edges<<<nb(E * 128, 256), 256, 0, stream>>>(h2, src, dst, ew, agg, E, 512, 7);
  graphconv_gemm<<<dim3(256 / 64, rowsBlk), 256, 0, stream>>>(
      agg, h2, Wl3, Wr3, bl3, bng, bnb, bnm, bnv, h3, N, 512, 256, 0);

  // ---- global mean pool + head
  zero_f32<<<nb(64 * 256 + 64, 256), 256, 0, stream>>>(gsum, 64 * 256 + 64);
  pool_sum<<<nb(N * 64, 256), 256, 0, stream>>>(h3, batch, gsum, N);
  pool_cnt<<<nb(N, 256), 256, 0, stream>>>(batch, gcnt, N);
  head_mlp<<<64, 64, 0, stream>>>(gsum, gcnt, W4, b4, W5, b5, (float*)d_out);
}